// constrained_network_44968307589559
// MI455X (gfx1250) — compile-verified
//
#include <hip/hip_runtime.h>
#include <hip/hip_bf16.h>
#include <math.h>

typedef __attribute__((ext_vector_type(2))) float v2f;
typedef __attribute__((ext_vector_type(8))) float v8f;

#define NSC 64
#define NVC 64
#define EMBC 32
#define NBC 8
#define RHC 16
#define CUTC 3.5f
#define HSTEP 0.1f
#define INVDEG 0.17677669529663687f   // 1/sqrt(32)
#define PIF 3.14159265358979323846f
#define BESSC 2.1380899352993947f     // sqrt(2/3.5)*sqrt(8)
#define SQRT3F 1.7320508075688772f

__device__ __forceinline__ float sigm(float x) { return 1.f / (1.f + __expf(-x)); }

// ---------------------------------------------------------------------------
// QR prep: Gram-Schmidt on the two columns of Ku.T (Ku is (2,64) row-major).
// Q stored row-major (64,2): Q[h*2+c]. (Sign convention: positive R diagonal.)
// ---------------------------------------------------------------------------
__global__ void qr_prep(const float* __restrict__ Ku, float* __restrict__ Qm) {
    if (threadIdx.x != 0 || blockIdx.x != 0) return;
    float n0 = 0.f;
    for (int h = 0; h < NVC; ++h) n0 += Ku[h] * Ku[h];
    float inv0 = rsqrtf(fmaxf(n0, 1e-30f));
    float r01 = 0.f;
    for (int h = 0; h < NVC; ++h) r01 += (Ku[h] * inv0) * Ku[NVC + h];
    float n1 = 0.f;
    for (int h = 0; h < NVC; ++h) {
        float u = Ku[NVC + h] - r01 * (Ku[h] * inv0);
        n1 += u * u;
    }
    float inv1 = rsqrtf(fmaxf(n1, 1e-30f));
    for (int h = 0; h < NVC; ++h) {
        float q0 = Ku[h] * inv0;
        float q1 = (Ku[NVC + h] - r01 * q0) * inv1;
        Qm[h * 2 + 0] = q0;
        Qm[h * 2 + 1] = q1;
    }
}

// ---------------------------------------------------------------------------
// State init: y_s = 0; y_v[n,h,d] = x[n,d]*Q[h,0] + x[n,3+d]*Q[h,1];
// pos = x[:, :3].
// ---------------------------------------------------------------------------
__global__ void init_state(const float* __restrict__ x, const float* __restrict__ Qm,
                           float* __restrict__ y_s, float* __restrict__ y_v,
                           float* __restrict__ pos, int N_) {
    int idx = blockIdx.x * blockDim.x + threadIdx.x;
    if (idx >= N_ * NVC) return;
    int n = idx >> 6, h = idx & 63;
    y_s[idx] = 0.f;
    float q0 = Qm[h * 2 + 0], q1 = Qm[h * 2 + 1];
    const float* xp = x + (size_t)n * 6;
    float* yvp = y_v + (size_t)n * 192 + h * 3;
    yvp[0] = xp[0] * q0 + xp[3] * q1;
    yvp[1] = xp[1] * q0 + xp[4] * q1;
    yvp[2] = xp[2] * q0 + xp[5] * q1;
    if (h < 3) pos[(size_t)n * 3 + h] = xp[h];
}

__global__ void zero_kernel(float* __restrict__ p, long count) {
    long i = (long)blockIdx.x * blockDim.x + threadIdx.x;
    long stride = (long)gridDim.x * blockDim.x;
    for (; i < count; i += stride) p[i] = 0.f;
}

// ---------------------------------------------------------------------------
// Edge kernel: one edge per wave32. Lane owns channels {lane, lane+32} and
// radial-MLP outputs w[lane + 32*i], i=0..7 == {wA,wB,wC,wD} x 2 channels.
// 10 global f32 atomics per lane into the L2-resident agg buffers.
// ---------------------------------------------------------------------------
__global__ __launch_bounds__(256) void edge_kernel(
    const int* __restrict__ esrc, const int* __restrict__ edst,
    const float* __restrict__ pos, const float* __restrict__ y_s,
    const float* __restrict__ y_v,
    const float* __restrict__ Wr1, const float* __restrict__ br1,
    const float* __restrict__ Wr2, const float* __restrict__ br2,
    float* __restrict__ agg_s, float* __restrict__ agg_v, int E_) {
    __shared__ float sWr2[RHC * 256];
    __shared__ float sWr1[NBC * RHC];
    __shared__ float sbr1[RHC];
    __shared__ float sbr2[256];
    int t = threadIdx.x;
    for (int i = t; i < RHC * 256; i += 256) sWr2[i] = Wr2[i];
    if (t < NBC * RHC) sWr1[t] = Wr1[t];
    if (t < RHC) sbr1[t] = br1[t];
    sbr2[t] = br2[t];
    __syncthreads();

    int e = blockIdx.x * 8 + (t >> 5);
    if (e >= E_) return;
    int lane = t & 31;
    int src = esrc[e], dst = edst[e];

    __builtin_prefetch(y_v + (size_t)src * 192 + lane * 3, 0, 0);

    const float* ps = pos + (size_t)src * 3;
    const float* pd = pos + (size_t)dst * 3;
    float evx = ps[0] - pd[0], evy = ps[1] - pd[1], evz = ps[2] - pd[2];
    float l2 = evx * evx + evy * evy + evz * evz;
    float elen = sqrtf(fmaxf(l2, 1e-24f));
    float invl = 1.f / elen;
    float u = elen * (1.f / CUTC);
    float vv = 2.f * (u - 1.f);
    float cof;
    if (vv > 0.f)        cof = 0.f;
    else if (vv < -1.f)  cof = 1.f;
    else                 cof = 0.5f * (1.f - cosf(PIF * vv));
    float sc = SQRT3F * invl * cof;
    float eax = sc * evx, eay = sc * evy, eaz = sc * evz;

    // radial MLP: bess(8) -> silu hidden(16) -> w(256)
    float h[RHC];
    #pragma unroll
    for (int r = 0; r < RHC; ++r) h[r] = sbr1[r];
    float arg = PIF * elen * (1.f / CUTC);
    #pragma unroll
    for (int k = 0; k < NBC; ++k) {
        float b = BESSC * sinf(arg * (float)(k + 1)) * invl;
        #pragma unroll
        for (int r = 0; r < RHC; ++r) h[r] += b * sWr1[k * RHC + r];
    }
    #pragma unroll
    for (int r = 0; r < RHC; ++r) h[r] = h[r] * sigm(h[r]);
    float w[8];
    #pragma unroll
    for (int i = 0; i < 8; ++i) {
        int o = lane + 32 * i;
        float acc = sbr2[o];
        #pragma unroll
        for (int r = 0; r < RHC; ++r) acc += h[r] * sWr2[r * 256 + o];
        w[i] = acc;
    }

    #pragma unroll
    for (int j = 0; j < 2; ++j) {
        int c = lane + 32 * j;
        float wa = w[0 + j], wb = w[2 + j], wc = w[4 + j], wd = w[6 + j];
        float sv = y_s[(size_t)src * NSC + c];
        const float* vp = y_v + (size_t)src * 192 + c * 3;
        float vx = vp[0], vy = vp[1], vz = vp[2];
        float dot = vx * eax + vy * eay + vz * eaz;
        float crx = vy * eaz - vz * eay;
        float cry = vz * eax - vx * eaz;
        float crz = vx * eay - vy * eax;
        atomicAdd(&agg_s[(size_t)dst * 128 + c], INVDEG * wb * dot);
        atomicAdd(&agg_s[(size_t)dst * 128 + 64 + c], INVDEG * wd * dot);
        float* av = agg_v + (size_t)dst * 192 + c * 3;
        atomicAdd(&av[0], INVDEG * (wa * sv * eax + wc * crx));
        atomicAdd(&av[1], INVDEG * (wa * sv * eay + wc * cry));
        atomicAdd(&av[2], INVDEG * (wa * sv * eaz + wc * crz));
    }
}

// ---------------------------------------------------------------------------
// Node kernel: 64-node tile per block (256 threads = 8 waves).
// Phase A: sc_s = Z(64x2048) @ Wss(2048x128), wave = col-tile (8), x4 row-tiles.
// Phase B: sc_v = Zv_d(64x2048) @ Wsv(2048x64) for d=0..2; wave = (col,row-pair),
//          B fragment reused across 3 dims x 2 row tiles.
// A fragments built on the fly from LDS: Z[row, s*32+e] = y*(row,s) * a(row,e).
// f32 WMMA 16x16x4 keeps the reference's fp32 fidelity.
// ---------------------------------------------------------------------------
__global__ __launch_bounds__(256) void node_kernel(
    float* __restrict__ y_s, float* __restrict__ y_v,
    const int* __restrict__ node_attr, const float* __restrict__ embed,
    const float* __restrict__ Wss, const float* __restrict__ Wsv,
    const float* __restrict__ agg_s, const float* __restrict__ agg_v,
    const float* __restrict__ Qm, float* __restrict__ pos,
    float* __restrict__ outp, int N_, int last) {
    extern __shared__ float sm[];
    float* ys  = sm;              // 64*64
    float* ae  = sm + 4096;       // 64*32
    float* yv  = sm + 6144;       // 3*64*64 : [d*4096 + row*64 + h]
    float* scs = sm + 18432;      // 64*128
    float* scv = sm + 26624;      // 3*64*64 : [d*4096 + row*64 + u]
    float* xv6 = sm + 38912;      // 64*6

    int t = threadIdx.x;
    int base = blockIdx.x * 64;

    for (int i = t; i < 4096; i += 256) {
        int n = base + (i >> 6);
        ys[i] = (n < N_) ? y_s[(size_t)n * NSC + (i & 63)] : 0.f;
    }
    for (int i = t; i < 2048; i += 256) {
        int n = base + (i >> 5);
        ae[i] = (n < N_) ? embed[node_attr[n] * EMBC + (i & 31)] : 0.f;
    }
    for (int i = t; i < 12288; i += 256) {
        int dd = i >> 12;
        int rem = i & 4095;
        int n = base + (rem >> 6);
        yv[i] = (n < N_) ? y_v[(size_t)n * 192 + (rem & 63) * 3 + dd] : 0.f;
    }
    for (int i = t; i < 384; i += 256) xv6[i] = 0.f;
    __syncthreads();

    int wave = t >> 5, lane = t & 31;
    int lrow = lane & 15, khalf = lane >> 4;

    // ---- Phase A: sc_s ----
    {
        v8f acc[4] = {};
        int ocol = wave * 16 + lrow;
        const float* Bp = Wss + khalf * 2 * 128 + ocol;
        int ro64[4], ro32[4];
        #pragma unroll
        for (int rt = 0; rt < 4; ++rt) {
            ro64[rt] = (rt * 16 + lrow) * 64;
            ro32[rt] = (rt * 16 + lrow) * 32;
        }
        for (int kk = 0; kk < 2048; kk += 4) {
            v2f b;
            b.x = Bp[0];
            b.y = Bp[128];
            Bp += 512;
            int sidx = kk >> 5;
            int e0 = (kk & 31) + khalf * 2;
            #pragma unroll
            for (int rt = 0; rt < 4; ++rt) {
                float yss = ys[ro64[rt] + sidx];
                v2f a;
                a.x = yss * ae[ro32[rt] + e0];
                a.y = yss * ae[ro32[rt] + e0 + 1];
                acc[rt] = __builtin_amdgcn_wmma_f32_16x16x4_f32(
                    false, a, false, b, (short)0, acc[rt], false, false);
            }
        }
        #pragma unroll
        for (int rt = 0; rt < 4; ++rt)
            #pragma unroll
            for (int i = 0; i < 8; ++i)
                scs[(rt * 16 + khalf * 8 + i) * 128 + ocol] = acc[rt][i];
    }

    // ---- Phase B: sc_v (3 dims) ----
    {
        v8f acc[3][2] = {};
        int ucol = (wave & 3) * 16 + lrow;
        int rtb = (wave >> 2) * 2;
        const float* Bp = Wsv + khalf * 2 * 64 + ucol;
        int ro64[2], ro32[2];
        #pragma unroll
        for (int j = 0; j < 2; ++j) {
            ro64[j] = ((rtb + j) * 16 + lrow) * 64;
            ro32[j] = ((rtb + j) * 16 + lrow) * 32;
        }
        for (int kk = 0; kk < 2048; kk += 4) {
            v2f b;
            b.x = Bp[0];
            b.y = Bp[64];
            Bp += 256;
            int sidx = kk >> 5;
            int e0 = (kk & 31) + khalf * 2;
            #pragma unroll
            for (int j = 0; j < 2; ++j) {
                float a0 = ae[ro32[j] + e0];
                float a1 = ae[ro32[j] + e0 + 1];
                #pragma unroll
                for (int dd = 0; dd < 3; ++dd) {
                    float yvv = yv[dd * 4096 + ro64[j] + sidx];
                    v2f a;
                    a.x = yvv * a0;
                    a.y = yvv * a1;
                    acc[dd][j] = __builtin_amdgcn_wmma_f32_16x16x4_f32(
                        false, a, false, b, (short)0, acc[dd][j], false, false);
                }
            }
        }
        #pragma unroll
        for (int dd = 0; dd < 3; ++dd)
            #pragma unroll
            for (int j = 0; j < 2; ++j)
                #pragma unroll
                for (int i = 0; i < 8; ++i)
                    scv[dd * 4096 + ((rtb + j) * 16 + khalf * 8 + i) * 64 + ucol] =
                        acc[dd][j][i];
    }
    __syncthreads();

    // ---- Epilogue: fuse agg, activations, state update, y_v @ Q ----
    {
        int nloc = t >> 2;
        int n = base + nloc;
        int cb = (t & 3) * 16;
        if (n < N_) {
            float p[6] = {0.f, 0.f, 0.f, 0.f, 0.f, 0.f};
            for (int i = 0; i < 16; ++i) {
                int c = cb + i;
                float o1 = scs[nloc * 128 + c] + agg_s[(size_t)n * 128 + c];
                float o2 = scs[nloc * 128 + 64 + c] + agg_s[(size_t)n * 128 + 64 + c];
                float ysn = ys[nloc * 64 + c] + HSTEP * o1 * sigm(o1);
                y_s[(size_t)n * NSC + c] = ysn;
                float gate = sigm(o2);
                float qa = Qm[c * 2 + 0], qb = Qm[c * 2 + 1];
                #pragma unroll
                for (int dd = 0; dd < 3; ++dd) {
                    float ov = scv[dd * 4096 + nloc * 64 + c] +
                               agg_v[(size_t)n * 192 + c * 3 + dd];
                    float yvn = yv[dd * 4096 + nloc * 64 + c] + HSTEP * gate * ov;
                    y_v[(size_t)n * 192 + c * 3 + dd] = yvn;
                    p[dd] += yvn * qa;
                    p[3 + dd] += yvn * qb;
                }
            }
            for (int j2 = 0; j2 < 6; ++j2) atomicAdd(&xv6[nloc * 6 + j2], p[j2]);
        }
    }
    __syncthreads();
    if (t < 64) {
        int n = base + t;
        if (n < N_) {
            pos[(size_t)n * 3 + 0] = xv6[t * 6 + 0];
            pos[(size_t)n * 3 + 1] = xv6[t * 6 + 1];
            pos[(size_t)n * 3 + 2] = xv6[t * 6 + 2];
            if (last) {
                #pragma unroll
                for (int j2 = 0; j2 < 6; ++j2)
                    outp[(size_t)n * 6 + j2] = xv6[t * 6 + j2];
            }
        }
    }
}

// ---------------------------------------------------------------------------
extern "C" void kernel_launch(void* const* d_in, const int* in_sizes, int n_in,
                              void* d_out, int out_size, void* d_ws, size_t ws_size,
                              hipStream_t stream) {
    (void)n_in; (void)out_size; (void)ws_size;
    const float* x        = (const float*)d_in[0];
    const int* node_attr  = (const int*)d_in[2];
    const int* edge_src   = (const int*)d_in[3];
    const int* edge_dst   = (const int*)d_in[4];
    const float* embed    = (const float*)d_in[5];
    const float* Ku       = (const float*)d_in[6];
    const float* Wss      = (const float*)d_in[7];
    const float* Wsv      = (const float*)d_in[8];
    const float* Wr1      = (const float*)d_in[9];
    const float* br1      = (const float*)d_in[10];
    const float* Wr2      = (const float*)d_in[11];
    const float* br2      = (const float*)d_in[12];
    float* outp = (float*)d_out;

    int N_ = in_sizes[0] / 6;
    int E_ = in_sizes[3];

    float* ws = (float*)d_ws;
    float* Qm    = ws;                         // 128 (pad 256)
    float* y_s   = Qm + 256;                   // N*64
    float* y_v   = y_s + (size_t)N_ * 64;      // N*192
    float* pos   = y_v + (size_t)N_ * 192;     // N*3
    float* agg_s = pos + (size_t)N_ * 3;       // N*128
    float* agg_v = agg_s + (size_t)N_ * 128;   // N*192  (contiguous with agg_s)

    qr_prep<<<1, 1, 0, stream>>>(Ku, Qm);
    init_state<<<(N_ * 64 + 255) / 256, 256, 0, stream>>>(x, Qm, y_s, y_v, pos, N_);

    const size_t node_smem = 39296 * sizeof(float);  // ~154 KB of 320 KB WGP LDS
    for (int l = 0; l < 2; ++l) {
        zero_kernel<<<2048, 256, 0, stream>>>(agg_s, (long)N_ * 320);
        edge_kernel<<<(E_ + 7) / 8, 256, 0, stream>>>(
            edge_src, edge_dst, pos, y_s, y_v,
            Wr1 + l * 128, br1 + l * 16, Wr2 + l * 4096, br2 + l * 256,
            agg_s, agg_v, E_);
        node_kernel<<<(N_ + 63) / 64, 256, node_smem, stream>>>(
            y_s, y_v, node_attr, embed,
            Wss + (size_t)l * 262144, Wsv + (size_t)l * 131072,
            agg_s, agg_v, Qm, pos, outp, N_, (l == 1) ? 1 : 0);
    }
}